// GAT_77653008712210
// MI455X (gfx1250) — compile-verified
//
#include <hip/hip_runtime.h>
#include <cstddef>

// GAT 2-layer forward for MI455X (gfx1250, wave32).
// Pipeline per layer:
//   [gemm]  feat = h @ W^T         -> fp32 WMMA (v_wmma_f32_16x16x4_f32)
//   [score] el/er = <feat, al/ar>  -> per-(node,head) dot
//   [edge]  w = exp(leaky_relu(el[src]+er[dst])); agg[dst] += w*feat[src];
//           denom[dst] += w        -> wave-per-(edge,head), coalesced atomics
//   [node]  out = elu(agg/denom)   (layer2: mean over heads)
// Edge phase is the HBM-bound part (~2 GB total @ 23.3 TB/s ~ 90us floor);
// GEMMs are ~3.3 GFLOP and ride the WMMA pipe.

typedef float v2f  __attribute__((ext_vector_type(2)));
typedef float v8f  __attribute__((ext_vector_type(8)));

#define HEADS 4
#define DIMH  32
#define FDIM  128   // HEADS*DIMH == IN_DIM == 128

// ---------------------------------------------------------------------------
// GEMM: out[r, :] = X[r, :] @ W^T,  X:[nrows,128], W:[128,128] row-major [o,k]
// Block = 256 threads = 8 waves; each wave -> 16 rows x 128 cols.
// LDS holds W pair-packed so each lane's B fragment {B[k][n],B[k+1][n]} is a
// contiguous 8B ds_load_b64 into an even-aligned VGPR pair (no repack moves):
//   Wp[g*128 + n] = { W[n][2g], W[n][2g+1] },  g = 2*kk + hi, n = nt*16 + m.
// ---------------------------------------------------------------------------
__global__ __launch_bounds__(256) void gat_gemm128(
    const float* __restrict__ X, const float* __restrict__ W,
    float* __restrict__ out, int nrows)
{
    __shared__ v2f Wp[64 * FDIM];              // 64KB

    const int t = threadIdx.x;
    #pragma unroll
    for (int i = 0; i < 32; ++i) {             // cooperative pair-packed load
        const int idx = t + i * 256;           // idx over 8192 pairs
        const int g = idx >> 7;
        const int n = idx & 127;
        Wp[idx] = (v2f){W[n * FDIM + 2 * g], W[n * FDIM + 2 * g + 1]};
    }
    __syncthreads();

    const int lane = t & 31;
    const int wave = t >> 5;
    const int rowbase = blockIdx.x * 128 + wave * 16;
    const int m  = lane & 15;                  // row (A) / col (B,C) within tile
    const int hi = lane >> 4;                  // lane-group select

    int r = rowbase + m;
    r = (r < nrows) ? r : (nrows - 1);         // branchless clamp: EXEC stays full
    const float* __restrict__ xr = X + (size_t)r * FDIM + (hi << 1);

    v8f acc[8];
    #pragma unroll
    for (int nt = 0; nt < 8; ++nt)
        acc[nt] = (v8f){0.f, 0.f, 0.f, 0.f, 0.f, 0.f, 0.f, 0.f};

    // fp32 WMMA 16x16x4: A 16x4 (lane m, VGPR pair = K {kb,kb+1}),
    // B 4x16 (lane n), C 16x16 (VGPR i -> row i+8*hi, lane -> col m).
    const v2f* __restrict__ wl = Wp + hi * FDIM + m;
    for (int kk = 0; kk < 32; ++kk) {
        const v2f a = *(const v2f*)(xr + kk * 4);      // global_load_b64
        const v2f* __restrict__ wk = wl + kk * (2 * FDIM);
        #pragma unroll
        for (int nt = 0; nt < 8; ++nt) {
            const v2f b = wk[nt * 16];                 // ds_load_b64
            acc[nt] = __builtin_amdgcn_wmma_f32_16x16x4_f32(
                false, a, false, b, (short)0, acc[nt], false, false);
        }
    }

    #pragma unroll
    for (int i = 0; i < 8; ++i) {
        const int row = rowbase + i + 8 * hi;
        if (row < nrows) {
            float* __restrict__ o = out + (size_t)row * FDIM + m;
            #pragma unroll
            for (int nt = 0; nt < 8; ++nt)
                o[nt * 16] = acc[nt][i];
        }
    }
}

// ---------------------------------------------------------------------------
// el[n,h] = feat[n,h,:].al[h,:]; er likewise. One thread per (node,head).
// ---------------------------------------------------------------------------
__global__ void gat_scores(const float* __restrict__ feat,
                           const float* __restrict__ al,
                           const float* __restrict__ ar,
                           float* __restrict__ el, float* __restrict__ er,
                           int nh)
{
    const int i = blockIdx.x * blockDim.x + threadIdx.x;
    if (i >= nh) return;
    const int h = i & (HEADS - 1);
    const float* __restrict__ f = feat + (size_t)(i >> 2) * FDIM + h * DIMH;
    const float* __restrict__ a = al + h * DIMH;
    const float* __restrict__ b = ar + h * DIMH;
    float sl = 0.f, sr = 0.f;
    #pragma unroll
    for (int d = 0; d < DIMH; ++d) {
        const float v = f[d];
        sl += v * a[d];
        sr += v * b[d];
    }
    el[i] = sl;
    er[i] = sr;
}

// ---------------------------------------------------------------------------
// Edge phase. 128 threads per edge -> each wave owns one (edge, head):
// feat[src] reads and agg[dst] atomics are 128B-contiguous per wave.
// Max-free softmax (scores are O(+-4); exp cannot overflow in f32, and the
// max-shift cancels exactly in alpha = ee/denom).
// ---------------------------------------------------------------------------
__global__ __launch_bounds__(256) void gat_edge(
    const float* __restrict__ feat,
    const int* __restrict__ src,
    const int* __restrict__ dst,
    const float* __restrict__ el,
    const float* __restrict__ er,
    float* __restrict__ agg,
    float* __restrict__ denom, int nedges)
{
    const long long tid = (long long)blockIdx.x * blockDim.x + threadIdx.x;
    const long long e = tid >> 7;
    if (e >= nedges) return;
    const int r = (int)(tid & 127);            // h*32 + d
    const int h = r >> 5;
    const int s  = src[e];
    const int d_ = dst[e];
    float x = el[s * HEADS + h] + er[d_ * HEADS + h];
    x = (x > 0.f) ? x : 0.2f * x;              // leaky_relu
    const float w = expf(x);
    atomicAdd(&agg[(size_t)d_ * FDIM + r], w * feat[(size_t)s * FDIM + r]);
    if ((r & (DIMH - 1)) == 0)
        atomicAdd(&denom[d_ * HEADS + h], w);
}

// ---------------------------------------------------------------------------
// Layer-1 epilogue: h1 = elu(agg / denom), one thread per (node, h*32+d).
// ---------------------------------------------------------------------------
__global__ void gat_norm_elu(const float* __restrict__ agg,
                             const float* __restrict__ denom,
                             float* __restrict__ out, int n)
{
    const int i = blockIdx.x * blockDim.x + threadIdx.x;
    if (i >= n) return;
    const int node = i >> 7;
    const int h = (i >> 5) & (HEADS - 1);
    float dn = denom[node * HEADS + h];
    dn = (dn == 0.f) ? 1.f : dn;               // isolated node -> output 0
    const float v = agg[i] / dn;
    out[i] = (v > 0.f) ? v : expm1f(v);
}

// ---------------------------------------------------------------------------
// Layer-2 epilogue: out[n,d] = mean_h elu(agg[n,h,d]/denom[n,h]).
// ---------------------------------------------------------------------------
__global__ void gat_final(const float* __restrict__ agg,
                          const float* __restrict__ denom,
                          float* __restrict__ out, int n)
{
    const int i = blockIdx.x * blockDim.x + threadIdx.x;
    if (i >= n) return;
    const int node = i >> 5;
    const int d = i & (DIMH - 1);
    float s = 0.f;
    #pragma unroll
    for (int h = 0; h < HEADS; ++h) {
        float dn = denom[node * HEADS + h];
        dn = (dn == 0.f) ? 1.f : dn;
        const float v = agg[(size_t)node * FDIM + h * DIMH + d] / dn;
        s += (v > 0.f) ? v : expm1f(v);
    }
    out[i] = 0.25f * s;
}

// ---------------------------------------------------------------------------
// Host side. Inputs: x, src, dst, W1, al1, ar1, W2, al2, ar2.
// ---------------------------------------------------------------------------
extern "C" void kernel_launch(void* const* d_in, const int* in_sizes, int n_in,
                              void* d_out, int out_size, void* d_ws, size_t ws_size,
                              hipStream_t stream)
{
    const float* x   = (const float*)d_in[0];
    const int*   src = (const int*)  d_in[1];
    const int*   dst = (const int*)  d_in[2];
    const float* W1  = (const float*)d_in[3];
    const float* al1 = (const float*)d_in[4];
    const float* ar1 = (const float*)d_in[5];
    const float* W2  = (const float*)d_in[6];
    const float* al2 = (const float*)d_in[7];
    const float* ar2 = (const float*)d_in[8];

    const int N = in_sizes[0] / FDIM;
    const int E = in_sizes[1];

    // Workspace layout (floats): feat | agg | h1 | el | er | denom  (~79 MB)
    float* feat  = (float*)d_ws;
    float* agg   = feat + (size_t)N * FDIM;
    float* h1    = agg  + (size_t)N * FDIM;
    float* el    = h1   + (size_t)N * FDIM;
    float* er    = el   + (size_t)N * HEADS;
    float* denom = er   + (size_t)N * HEADS;

    const int gemmBlocks  = (N + 127) / 128;
    const int scoreBlocks = (N * HEADS + 255) / 256;
    const int edgeBlocks  = (int)(((long long)E * FDIM + 255) / 256);
    const int normBlocks  = (N * FDIM + 255) / 256;
    const int finalBlocks = (N * DIMH + 255) / 256;

    // ---- layer 1 ----
    gat_gemm128<<<gemmBlocks, 256, 0, stream>>>(x, W1, feat, N);
    gat_scores<<<scoreBlocks, 256, 0, stream>>>(feat, al1, ar1, el, er, N * HEADS);
    hipMemsetAsync(agg,   0, (size_t)N * FDIM  * sizeof(float), stream);
    hipMemsetAsync(denom, 0, (size_t)N * HEADS * sizeof(float), stream);
    gat_edge<<<edgeBlocks, 256, 0, stream>>>(feat, src, dst, el, er, agg, denom, E);
    gat_norm_elu<<<normBlocks, 256, 0, stream>>>(agg, denom, h1, N * FDIM);

    // ---- layer 2 ----
    gat_gemm128<<<gemmBlocks, 256, 0, stream>>>(h1, W2, feat, N);
    gat_scores<<<scoreBlocks, 256, 0, stream>>>(feat, al2, ar2, el, er, N * HEADS);
    hipMemsetAsync(agg,   0, (size_t)N * FDIM  * sizeof(float), stream);
    hipMemsetAsync(denom, 0, (size_t)N * HEADS * sizeof(float), stream);
    gat_edge<<<edgeBlocks, 256, 0, stream>>>(feat, src, dst, el, er, agg, denom, E);
    gat_final<<<finalBlocks, 256, 0, stream>>>(agg, denom, (float*)d_out, N * DIMH);
}